// PostProcess_33784212750559
// MI455X (gfx1250) — compile-verified
//
#include <hip/hip_runtime.h>
#include <hip/hip_bf16.h>
#include <math.h>

// ---------------------------------------------------------------------------
// Problem constants (from reference)
// ---------------------------------------------------------------------------
#define B_    256
#define Q_    1000
#define C_    91
#define N_    (Q_ * C_)        // 91000 logits per batch
#define K_    100              // TOPK_DET
#define NS_   4

// Flat output layout (floats), in reference return order
#define OFF_SCORES 0
#define OFF_LABELS (B_ * K_)                     // 25600
#define OFF_BOXES  (2 * B_ * K_)                 // 51200
#define OFF_AMOUNT (OFF_BOXES + B_ * K_ * 4)     // 153600
#define OFF_WB     (OFF_AMOUNT + B_ * K_ * 4)    // 256000
#define WB_V_SZ    (B_ * 3 * 3)                  // 2304
#define WB_B_SZ    (B_ * 3 * 3 * 4)              // 9216

// ---------------------------------------------------------------------------
// CDNA5 async global->LDS path (gfx1250). Guarded so compile never breaks.
// Builtin signature (from hipcc diagnostic): first arg is
//   'int __vector(4) __device__ *'  i.e. AS1 pointer to a 16-byte int vector.
// ---------------------------------------------------------------------------
#if defined(__gfx1250__) &&                                              \
    __has_builtin(__builtin_amdgcn_global_load_async_to_lds_b128) &&      \
    __has_builtin(__builtin_amdgcn_s_wait_asynccnt)
#define USE_ASYNC_LDS 1
typedef int v4i_ __attribute__((vector_size(16)));
typedef __attribute__((address_space(1))) v4i_* gptr_b128;
typedef __attribute__((address_space(3))) v4i_* lptr_b128;
#else
#define USE_ASYNC_LDS 0
#endif

// ---------------------------------------------------------------------------
// Order-preserving float<->uint key mapping (larger key == larger float)
// ---------------------------------------------------------------------------
__device__ __forceinline__ unsigned fkey(float f) {
  unsigned u = __float_as_uint(f);
  return (u & 0x80000000u) ? ~u : (u | 0x80000000u);
}
__device__ __forceinline__ float keyToFloat(unsigned k) {
  unsigned u = (k & 0x80000000u) ? (k & 0x7FFFFFFFu) : ~k;
  return __uint_as_float(u);
}

// ---------------------------------------------------------------------------
// Kernel 1: per-batch top-100 over 91000 logits via 4-round radix select,
// then collect + bitonic sort, then scores/labels/boxes outputs.
// One 256-thread (8-wave) block per batch.
// ---------------------------------------------------------------------------
__global__ __launch_bounds__(256) void topk_det_kernel(
    const float* __restrict__ logits, const float* __restrict__ boxes_in,
    const float* __restrict__ tsizes, float* __restrict__ out,
    int* __restrict__ ws_q)
{
  const int b   = blockIdx.x;
  const int tid = threadIdx.x;
  const float* base = logits + (size_t)b * N_;

  __shared__ unsigned hist[256];
  __shared__ unsigned s_prefix, s_kRem, s_cnt;
  __shared__ unsigned long long skey[256];
#if USE_ASYNC_LDS
  __shared__ __align__(16) float tile[1024];   // 4 KB staging tile
#endif

  // ---- 4-round radix select: find exact key T of the 100th largest ----
  unsigned prefix = 0, kRem = K_;
  for (int round = 0; round < 4; ++round) {
    const int shift = 24 - 8 * round;
    hist[tid] = 0;
    __syncthreads();

    if (round == 0) {
#if USE_ASYNC_LDS
      const int nFull = (N_ / 1024) * 1024;   // 90112
      for (int t0 = 0; t0 < nFull; t0 += 1024) {
        // CDNA5 async copy: each lane moves 16B global -> LDS (ASYNCcnt)
        __builtin_amdgcn_global_load_async_to_lds_b128(
            (gptr_b128)(base + t0 + tid * 4),
            (lptr_b128)(&tile[tid * 4]), 0, 0);
        __builtin_amdgcn_s_wait_asynccnt(0);
        __syncthreads();
#pragma unroll
        for (int u = 0; u < 4; ++u) {
          unsigned key = fkey(tile[tid * 4 + u]);
          atomicAdd(&hist[key >> 24], 1u);
        }
        __syncthreads();   // tile fully consumed before next async overwrite
      }
      for (int i = nFull + tid; i < N_; i += 256) {
        unsigned key = fkey(base[i]);
        atomicAdd(&hist[key >> 24], 1u);
      }
#else
      for (int i = tid; i < N_; i += 256) {
        unsigned key = fkey(base[i]);
        atomicAdd(&hist[key >> 24], 1u);
      }
#endif
    } else {
      const unsigned pmask = 0xFFFFFFFFu << (shift + 8);
      for (int i = tid; i < N_; i += 256) {
        __builtin_prefetch(base + i + 2048, 0, 1);   // global_prefetch_b8
        unsigned key = fkey(base[i]);
        if ((key & pmask) == prefix)
          atomicAdd(&hist[(key >> shift) & 0xFFu], 1u);
      }
    }
    __syncthreads();

    if (tid == 0) {
      unsigned acc = 0;
      int bin = 255;
      while (bin > 0 && acc + hist[bin] < kRem) { acc += hist[bin]; --bin; }
      s_prefix = prefix | ((unsigned)bin << shift);
      s_kRem   = kRem - acc;
    }
    __syncthreads();
    prefix = s_prefix;
    kRem   = s_kRem;
    __syncthreads();
  }
  const unsigned T = prefix;   // exact key of the 100th-largest element

  // ---- collect all elements with key >= T (>= 100, ~100 in practice) ----
  if (tid == 0) s_cnt = 0;
  skey[tid] = 0xFFFFFFFFFFFFFFFFull;   // pad: sorts last (key=0, idx=max)
  __syncthreads();
  for (int i = tid; i < N_; i += 256) {
    unsigned key = fkey(base[i]);
    if (key >= T) {
      unsigned p = atomicAdd(&s_cnt, 1u);
      if (p < 256u)
        skey[p] = ((unsigned long long)(~key) << 32) | (unsigned)i;
    }
  }
  __syncthreads();

  // ---- bitonic sort 256 entries ascending on (~key, idx)
  //      == (value desc, index asc), matching jax.lax.top_k ----
  for (unsigned kk = 2; kk <= 256; kk <<= 1) {
    for (unsigned j = kk >> 1; j > 0; j >>= 1) {
      unsigned i = (unsigned)tid, ixj = i ^ j;
      if (ixj > i) {
        unsigned long long a = skey[i], c = skey[ixj];
        bool asc = ((i & kk) == 0);
        if ((a > c) == asc) { skey[i] = c; skey[ixj] = a; }
      }
      __syncthreads();
    }
  }

  // ---- emit scores / labels / scaled boxes; stash q-indices for kernel 2 --
  if (tid < K_) {
    unsigned long long sk = skey[tid];
    unsigned idx = (unsigned)sk;
    unsigned key = ~(unsigned)(sk >> 32);
    float lg = keyToFloat(key);
    float score = 1.0f / (1.0f + expf(-lg));

    unsigned q = idx / C_;
    unsigned c = idx % C_;

    out[OFF_SCORES + b * K_ + tid] = score;
    out[OFF_LABELS + b * K_ + tid] = (float)c;
    ws_q[b * K_ + tid] = (int)q;

    const float* bx = boxes_in + ((size_t)b * Q_ + q) * 4;
    float cx = bx[0], cy = bx[1], w = bx[2], h = bx[3];
    float ih = tsizes[b * 2 + 0], iw = tsizes[b * 2 + 1];
    float* o = out + OFF_BOXES + ((size_t)b * K_ + tid) * 4;
    o[0] = (cx - 0.5f * w) * iw;
    o[1] = (cy - 0.5f * h) * ih;
    o[2] = (cx + 0.5f * w) * iw;
    o[3] = (cy + 0.5f * h) * ih;
  }
}

// ---------------------------------------------------------------------------
// Kernel 2: amount_sel = amount_score[:, topk_boxes[0], :]
// (uses BATCH 0's q-indices for every batch, per the reference)
// ---------------------------------------------------------------------------
__global__ __launch_bounds__(128) void amount_kernel(
    const float* __restrict__ amount, const int* __restrict__ ws_q,
    float* __restrict__ out)
{
  const int b = blockIdx.x;
  for (int e = threadIdx.x; e < K_ * 4; e += 128) {
    int j = e >> 2, a = e & 3;
    int q = ws_q[j];   // batch-0 indices
    out[OFF_AMOUNT + b * K_ * 4 + e] = amount[((size_t)b * Q_ + q) * 4 + a];
  }
}

// ---------------------------------------------------------------------------
// Kernel 3: _weighted_bbox for hs/enc/pca weights.
// One wave32 block per (tensor t, slice s in 1..3, batch b) row of 1000.
// ---------------------------------------------------------------------------
__global__ __launch_bounds__(32) void wbbox_kernel(
    const float* __restrict__ hs, const float* __restrict__ enc,
    const float* __restrict__ pb, const float* __restrict__ boxes_in,
    const float* __restrict__ tsizes, float* __restrict__ out)
{
  const int lane = threadIdx.x;
  const int s = blockIdx.x % 3;       // slice s+1 of NS
  const int t = blockIdx.x / 3;       // 0=hs, 1=enc, 2=pca
  const int b = blockIdx.y;

  const float* W = (t == 0) ? hs : (t == 1) ? enc : pb;
  const float* row = W + ((size_t)b * NS_ + (s + 1)) * Q_;

  // per-lane top-3 over the lane's stripe (strict > keeps lowest index)
  float v[3] = {-INFINITY, -INFINITY, -INFINITY};
  int  ix[3] = {0x7FFFFFFF, 0x7FFFFFFF, 0x7FFFFFFF};
  for (int i = lane; i < Q_; i += 32) {
    __builtin_prefetch(row + i + 256, 0, 1);
    float val = row[i];
    if (val > v[0])      { v[2]=v[1]; ix[2]=ix[1]; v[1]=v[0]; ix[1]=ix[0]; v[0]=val; ix[0]=i; }
    else if (val > v[1]) { v[2]=v[1]; ix[2]=ix[1]; v[1]=val;  ix[1]=i; }
    else if (val > v[2]) { v[2]=val;  ix[2]=i; }
  }

  __shared__ float sv[96];
  __shared__ int   si[96];
  sv[lane*3+0] = v[0]; sv[lane*3+1] = v[1]; sv[lane*3+2] = v[2];
  si[lane*3+0] = ix[0]; si[lane*3+1] = ix[1]; si[lane*3+2] = ix[2];
  __syncthreads();

  if (lane == 0) {
    float bv[3] = {-INFINITY, -INFINITY, -INFINITY};
    int   bi[3] = {0x7FFFFFFF, 0x7FFFFFFF, 0x7FFFFFFF};
    for (int n = 0; n < 96; ++n) {
      float val = sv[n];
      int   i   = si[n];
      bool a0 = (val > bv[0]) || (val == bv[0] && i < bi[0]);
      bool a1 = (val > bv[1]) || (val == bv[1] && i < bi[1]);
      bool a2 = (val > bv[2]) || (val == bv[2] && i < bi[2]);
      if (a0)      { bv[2]=bv[1]; bi[2]=bi[1]; bv[1]=bv[0]; bi[1]=bi[0]; bv[0]=val; bi[0]=i; }
      else if (a1) { bv[2]=bv[1]; bi[2]=bi[1]; bv[1]=val;  bi[1]=i; }
      else if (a2) { bv[2]=val;  bi[2]=i; }
    }

    const int vOff = OFF_WB + t * (WB_V_SZ + WB_B_SZ);
    const int bOff = vOff + WB_V_SZ;
    float ih = tsizes[b * 2 + 0], iw = tsizes[b * 2 + 1];
    for (int k = 0; k < 3; ++k) {
      out[vOff + (b * 3 + s) * 3 + k] = bv[k];
      int q = bi[k];
      const float* bx = boxes_in + ((size_t)b * Q_ + q) * 4;
      float cx = bx[0], cy = bx[1], w = bx[2], h = bx[3];
      float* o = out + bOff + ((b * 3 + s) * 3 + k) * 4;
      o[0] = (cx - 0.5f * w) * iw;
      o[1] = (cy - 0.5f * h) * ih;
      o[2] = (cx + 0.5f * w) * iw;
      o[3] = (cy + 0.5f * h) * ih;
    }
  }
}

// ---------------------------------------------------------------------------
// Launch
// ---------------------------------------------------------------------------
extern "C" void kernel_launch(void* const* d_in, const int* in_sizes, int n_in,
                              void* d_out, int out_size, void* d_ws, size_t ws_size,
                              hipStream_t stream) {
  const float* logits = (const float*)d_in[0];   // (B,Q,C)
  const float* boxes  = (const float*)d_in[1];   // (B,Q,4)
  const float* ts     = (const float*)d_in[2];   // (B,2) = (h,w)
  const float* amount = (const float*)d_in[3];   // (B,Q,4)
  const float* hs     = (const float*)d_in[4];   // (B,NS,Q)
  const float* enc    = (const float*)d_in[5];
  const float* pb     = (const float*)d_in[6];
  float* out = (float*)d_out;
  int*   ws_q = (int*)d_ws;                      // B*K ints (102400 B)

  topk_det_kernel<<<B_, 256, 0, stream>>>(logits, boxes, ts, out, ws_q);
  amount_kernel<<<B_, 128, 0, stream>>>(amount, ws_q, out);
  wbbox_kernel<<<dim3(9, B_), 32, 0, stream>>>(hs, enc, pb, boxes, ts, out);
}